// MultiRelationGNNLayer_67817533604356
// MI455X (gfx1250) — compile-verified
//
#include <hip/hip_runtime.h>
#include <hip/hip_bf16.h>

typedef __attribute__((ext_vector_type(2))) float v2f;
typedef __attribute__((ext_vector_type(8))) float v8f;

#define DIM   128
#define NREL  3
#define KTOT  512   // 128 (x @ lin_w^T) + 3*128 (A_r @ W_r)
#define ACOLS 384   // NREL * DIM
#define TILEM 32    // rows per block

// ---------------------------------------------------------------------------
// Zero the accumulator region of the workspace (A: N*384 floats, deg: N floats)
// ---------------------------------------------------------------------------
__global__ void rgcn_zero_kernel(float* __restrict__ p, long n) {
    long i = (long)blockIdx.x * blockDim.x + threadIdx.x;
    long stride = (long)gridDim.x * blockDim.x;
    for (; i < n; i += stride) p[i] = 0.0f;
}

// ---------------------------------------------------------------------------
// Build pair-interleaved combined weight WcP:
//   logical Wc[512][128]:
//     k <  128 : Wc[k][j] = lin_w[j][k]         (transpose for x @ lin_w^T)
//     k >= 128 : Wc[k][j] = relation_weights flat[(k-128)*128 + j]
//   stored as float2 pairs: WcP[(k>>1)*128 + j] = { Wc[k][j], Wc[k+1][j] }
//   so a B-fragment is one 8-byte load.
// ---------------------------------------------------------------------------
__global__ void rgcn_fill_wc_kernel(const float* __restrict__ lin_w,
                                    const float* __restrict__ rel_w,
                                    float* __restrict__ WcP) {
    int idx = blockIdx.x * blockDim.x + threadIdx.x;   // 0 .. 512*128-1
    int k = idx >> 7;
    int j = idx & 127;
    float v = (k < DIM) ? lin_w[j * DIM + k] : rel_w[(k - DIM) * DIM + j];
    WcP[(((k >> 1) * DIM + j) << 1) | (k & 1)] = v;
}

// ---------------------------------------------------------------------------
// Edge phase: one wave32 per edge. Gather x[src] (float4 per lane) and
// atomically accumulate into A[dst][rel*128 + c]; lane 0 bumps deg[dst].
// Pure L2 bandwidth: x (25.6 MB) and A (76.8 MB) are both L2-resident.
// ---------------------------------------------------------------------------
__global__ __launch_bounds__(256)
void rgcn_edge_kernel(const float* __restrict__ x,
                      const long long* __restrict__ edge_index, // [2][E]
                      const long long* __restrict__ edge_type,  // [E]
                      float* __restrict__ A,                    // [N][384]
                      float* __restrict__ deg,                  // [N]
                      int E) {
    int w    = (blockIdx.x * 256 + threadIdx.x) >> 5;  // edge id (uniform per wave)
    int lane = threadIdx.x & 31;
    if (w >= E) return;

    int src = (int)edge_index[w];
    int dst = (int)edge_index[E + w];
    int rel = (int)edge_type[w];

    const float4* xs = (const float4*)(x + (long)src * DIM);
    float4 v = xs[lane];                               // 4 floats per lane

    float* dp = A + (long)dst * ACOLS + rel * DIM + lane * 4;
    unsafeAtomicAdd(dp + 0, v.x);
    unsafeAtomicAdd(dp + 1, v.y);
    unsafeAtomicAdd(dp + 2, v.z);
    unsafeAtomicAdd(dp + 3, v.w);
    if (lane == 0) unsafeAtomicAdd(deg + dst, 1.0f);
}

// ---------------------------------------------------------------------------
// Node phase: fused  h = [x | A/deg] @ Wc + lin_b  followed by LayerNorm.
// Block = 256 threads = 8 wave32s; block owns 32 rows; each wave computes a
// 32x16 output stripe as two 16x16 accumulators sharing one B-fragment per
// V_WMMA_F32_16X16X4_F32 step (one b64 B-load feeds two WMMAs).
// ---------------------------------------------------------------------------
__global__ __launch_bounds__(256)
void rgcn_gemm_ln_kernel(const float* __restrict__ x,
                         const float* __restrict__ A,      // [N][384]
                         const float* __restrict__ deg,    // [N]
                         const float* __restrict__ WcP,    // paired [256][128] f32x2
                         const float* __restrict__ lin_b,
                         const float* __restrict__ gamma,
                         const float* __restrict__ beta,
                         float* __restrict__ out,          // [N][128]
                         int N) {
    __shared__ float rowbuf[TILEM * KTOT];   // 64 KB: 32 combined rows of 512

    const int n0   = blockIdx.x * TILEM;
    const int t    = threadIdx.x;
    const int lane = t & 31;
    const int wave = t >> 5;

    // Stage 32 x 512 combined rows (float4 granularity):
    //   cols 0..127 = x[n], cols 128..511 = A[n]/max(deg,1)
    #pragma unroll
    for (int i = 0; i < 16; ++i) {
        int idx4 = t + i * 256;          // 0 .. 4095 (float4 index)
        int row  = idx4 >> 7;            // /128 float4s per row
        int c4   = idx4 & 127;
        int n    = n0 + row;
        float4 v = make_float4(0.f, 0.f, 0.f, 0.f);
        if (n < N) {
            if (c4 < DIM / 4) {
                v = ((const float4*)(x + (long)n * DIM))[c4];
            } else {
                float4 a = ((const float4*)(A + (long)n * ACOLS))[c4 - DIM / 4];
                float inv = 1.0f / fmaxf(deg[n], 1.0f);
                v.x = a.x * inv; v.y = a.y * inv; v.z = a.z * inv; v.w = a.w * inv;
            }
        }
        ((float4*)rowbuf)[idx4] = v;
    }
    __syncthreads();

    // WMMA chain: this wave's 32x16 stripe at columns [j0, j0+16)
    const int j0    = wave * 16;
    const int msub  = lane & 15;          // A-matrix row / D-matrix column index
    const int khalf = (lane >> 4) << 1;   // 0 for lanes 0-15, 2 for lanes 16-31
    const int j     = j0 + msub;
    const v2f* Wp   = (const v2f*)WcP;

    v8f c0 = {};
    v8f c1 = {};
    #pragma unroll 4
    for (int kk = 0; kk < KTOT; kk += 4) {
        int kb = kk + khalf;
        // A-frags (16x4 f32): lane L holds rowbuf[M][kb], rowbuf[M][kb+1]
        v2f a0 = *(const v2f*)(rowbuf + msub * KTOT + kb);
        v2f a1 = *(const v2f*)(rowbuf + (msub + 16) * KTOT + kb);
        // Shared B-frag (4x16 f32): one b64 load of {Wc[kb][j], Wc[kb+1][j]}
        v2f b = Wp[(kb >> 1) * DIM + j];
        c0 = __builtin_amdgcn_wmma_f32_16x16x4_f32(
                false, a0, false, b, (short)0, c0, false, false);
        c1 = __builtin_amdgcn_wmma_f32_16x16x4_f32(
                false, a1, false, b, (short)0, c1, false, false);
    }

    float bias = lin_b[j];
    __syncthreads();                       // done reading rowbuf; reuse for h
    float* hbuf = rowbuf;                  // [32][128]
    #pragma unroll
    for (int v = 0; v < 8; ++v) {
        int m = v + ((lane >> 4) << 3);    // C/D layout: VGPR v -> M=v or v+8
        hbuf[m * DIM + j]          = c0[v] + bias;
        hbuf[(m + 16) * DIM + j]   = c1[v] + bias;
    }
    // LN scratch lives in the dead part of rowbuf just past hbuf
    float* red   = rowbuf + TILEM * DIM;       // [2][32][8]
    float* stats = red + 2 * TILEM * 8;        // [2][32]
    __syncthreads();

    // LayerNorm: 8 threads per row, 16 elements each
    int row = t >> 3;                      // 0..31
    int sub = t & 7;                       // 0..7
    float s = 0.0f, s2 = 0.0f;
    #pragma unroll
    for (int i = 0; i < 16; ++i) {
        float v = hbuf[row * DIM + sub + i * 8];
        s += v; s2 += v * v;
    }
    red[row * 8 + sub]              = s;
    red[TILEM * 8 + row * 8 + sub]  = s2;
    __syncthreads();
    if (sub == 0) {
        float ts = 0.0f, ts2 = 0.0f;
        #pragma unroll
        for (int i = 0; i < 8; ++i) {
            ts  += red[row * 8 + i];
            ts2 += red[TILEM * 8 + row * 8 + i];
        }
        float mu  = ts * (1.0f / DIM);
        float var = ts2 * (1.0f / DIM) - mu * mu;
        stats[row]         = mu;
        stats[TILEM + row] = rsqrtf(var + 1e-5f);
    }
    __syncthreads();

    int n = n0 + row;
    if (n < N) {
        float mu = stats[row], rs = stats[TILEM + row];
        #pragma unroll
        for (int i = 0; i < 16; ++i) {
            int col = sub + i * 8;
            float v = hbuf[row * DIM + col];
            out[(long)n * DIM + col] = gamma[col] * (v - mu) * rs + beta[col];
        }
    }
}

// ---------------------------------------------------------------------------
extern "C" void kernel_launch(void* const* d_in, const int* in_sizes, int n_in,
                              void* d_out, int out_size, void* d_ws, size_t ws_size,
                              hipStream_t stream) {
    const float*     x     = (const float*)d_in[0];
    const long long* eidx  = (const long long*)d_in[1];
    const long long* etype = (const long long*)d_in[2];
    const float*     rel_w = (const float*)d_in[3];   // [3][128][128]
    const float*     lin_w = (const float*)d_in[4];   // [128][128]
    const float*     lin_b = (const float*)d_in[5];
    const float*     gamma = (const float*)d_in[6];
    const float*     beta  = (const float*)d_in[7];

    const int N = in_sizes[0] / DIM;
    const int E = in_sizes[1] / 2;

    // workspace layout: A [N*384] | deg [N] | WcP [512*128]
    float* A   = (float*)d_ws;
    float* deg = A + (long)N * ACOLS;
    float* WcP = deg + N;

    long zn = (long)N * (ACOLS + 1);
    rgcn_zero_kernel<<<4096, 256, 0, stream>>>(A, zn);
    rgcn_fill_wc_kernel<<<(KTOT * DIM) / 256, 256, 0, stream>>>(lin_w, rel_w, WcP);
    rgcn_edge_kernel<<<(E + 7) / 8, 256, 0, stream>>>(x, eidx, etype, A, deg, E);
    rgcn_gemm_ln_kernel<<<(N + TILEM - 1) / TILEM, 256, 0, stream>>>(
        x, A, deg, WcP, lin_b, gamma, beta, (float*)d_out, N);
}